// WordDropout_53154515255825
// MI455X (gfx1250) — compile-verified
//
#include <hip/hip_runtime.h>
#include <stdint.h>

// Word dropout: out[i] = mask[i] ? UNK : x[i]
// Pure streaming elementwise op: 9 B/elem HBM traffic, ~0 FLOPs ->
// bandwidth-bound (~302 MB/call, ~13 us floor at 23.3 TB/s).
// gfx1250 tuning:
//  - 128-bit global loads/stores with non-temporal hints (288 MB working
//    set > 192 MB L2, zero reuse -> stream, don't pollute).
//  - 4-way MLP per wave32 lane; the 4 strided base pointers are computed
//    on the HOST and passed as kernel args, so every memory op is
//    "saddr + v0 (scale_offset)" with no per-lane 64-bit address math.
//  - Branch-free per-component select -> v_cndmask_b32.

#define UNK_TOKEN 3.0f

typedef __attribute__((ext_vector_type(4))) float v4f;

__device__ __forceinline__ v4f masked_fill4(v4f v, uint32_t m4) {
    // m4 holds 4 mask bytes (jax bool = 1 byte/elem). Nonzero byte -> UNK.
    v4f r;
    r.x = (m4 & 0x000000FFu) ? UNK_TOKEN : v.x;
    r.y = (m4 & 0x0000FF00u) ? UNK_TOKEN : v.y;
    r.z = (m4 & 0x00FF0000u) ? UNK_TOKEN : v.z;
    r.w = (m4 & 0xFF000000u) ? UNK_TOKEN : v.w;
    return r;
}

__global__ __launch_bounds__(256) void word_dropout_vec_kernel(
    const v4f* __restrict__ x0, const v4f* __restrict__ x1,
    const v4f* __restrict__ x2, const v4f* __restrict__ x3,
    const uint32_t* __restrict__ m0, const uint32_t* __restrict__ m1,
    const uint32_t* __restrict__ m2, const uint32_t* __restrict__ m3,
    v4f* __restrict__ o0, v4f* __restrict__ o1,
    v4f* __restrict__ o2, v4f* __restrict__ o3,
    int nvec, int nt)
{
    int i = (int)(blockIdx.x * blockDim.x + threadIdx.x);

    // Fast path (taken by every thread for the reference shape):
    // 8 independent NT loads in flight, all addressed as saddr_k + i.
    if (i + 3 * nt < nvec) {
        v4f v0 = __builtin_nontemporal_load(x0 + i);
        v4f v1 = __builtin_nontemporal_load(x1 + i);
        v4f v2 = __builtin_nontemporal_load(x2 + i);
        v4f v3 = __builtin_nontemporal_load(x3 + i);
        uint32_t a0 = __builtin_nontemporal_load(m0 + i);
        uint32_t a1 = __builtin_nontemporal_load(m1 + i);
        uint32_t a2 = __builtin_nontemporal_load(m2 + i);
        uint32_t a3 = __builtin_nontemporal_load(m3 + i);

        __builtin_nontemporal_store(masked_fill4(v0, a0), o0 + i);
        __builtin_nontemporal_store(masked_fill4(v1, a1), o1 + i);
        __builtin_nontemporal_store(masked_fill4(v2, a2), o2 + i);
        __builtin_nontemporal_store(masked_fill4(v3, a3), o3 + i);
        return;
    }

    // Boundary threads: strided loop over remaining chunks (grid sizing
    // guarantees 4*nt >= nvec, so fast path + this loop cover exactly).
    for (; i < nvec; i += nt) {
        v4f v = __builtin_nontemporal_load(x0 + i);
        uint32_t m = __builtin_nontemporal_load(m0 + i);
        __builtin_nontemporal_store(masked_fill4(v, m), o0 + i);
    }
}

// Scalar tail for n % 4 elements (empty for 4096*8192, kept for generality).
__global__ void word_dropout_tail_kernel(
    const float* __restrict__ x,
    const unsigned char* __restrict__ mask,
    float* __restrict__ out,
    int start, int n)
{
    int i = start + (int)(blockIdx.x * blockDim.x + threadIdx.x);
    if (i < n) {
        out[i] = mask[i] ? UNK_TOKEN : x[i];
    }
}

extern "C" void kernel_launch(void* const* d_in, const int* in_sizes, int n_in,
                              void* d_out, int out_size, void* d_ws, size_t ws_size,
                              hipStream_t stream) {
    (void)n_in; (void)out_size; (void)d_ws; (void)ws_size;

    const float* x = (const float*)d_in[0];
    const unsigned char* mask = (const unsigned char*)d_in[1];  // jax bool: 1 byte/elem
    float* out = (float*)d_out;

    const int n = in_sizes[0];
    const int nvec = n >> 2;    // float4 / packed-4-mask-byte chunks
    const int rem = n & 3;

    const int threads = 256;                 // 8 wave32 per block
    const int per_block = threads * 4;       // 4 vec4s per lane (ILP)
    if (nvec > 0) {
        int blocks = (nvec + per_block - 1) / per_block;  // 8192 for 4096x8192
        int nt = blocks * threads;                        // total threads

        const v4f* x4 = (const v4f*)x;
        const uint32_t* m4 = (const uint32_t*)mask;
        v4f* o4 = (v4f*)out;

        word_dropout_vec_kernel<<<blocks, threads, 0, stream>>>(
            x4, x4 + nt, x4 + 2 * nt, x4 + 3 * nt,
            m4, m4 + nt, m4 + 2 * nt, m4 + 3 * nt,
            o4, o4 + nt, o4 + 2 * nt, o4 + 3 * nt,
            nvec, nt);
    }
    if (rem > 0) {
        word_dropout_tail_kernel<<<1, 4, 0, stream>>>(x, mask, out, nvec << 2, n);
    }
}